// BeamDelay2AntFreq_3D_42984032698896
// MI455X (gfx1250) — compile-verified
//
#include <hip/hip_runtime.h>
#include <hip/hip_bf16.h>

typedef __attribute__((ext_vector_type(2))) float v2f;
typedef __attribute__((ext_vector_type(4))) float v4f;
typedef __attribute__((ext_vector_type(8))) float v8f;

#define WMMA_F32X4(a, b, c) \
  __builtin_amdgcn_wmma_f32_16x16x4_f32(false, (a), false, (b), (short)0, (c), false, false)

// ---- d_ws float layout: MR@0[1024] MI@1024[1024] BR@2048[2304] BI@4352[2304] = 6656 floats ----
#define TAB_FLOATS 6656

__global__ __launch_bounds__(256) void build_tables_kernel(float* __restrict__ ws) {
  const int idx = blockIdx.x * 256 + threadIdx.x;
  const float TWO_PI = 6.283185307179586f;
  if (idx < 1024) {  // M = A8 (x) A4 : mag 1/sqrt(32), sign (-1)^(h'+v'), exact phase mod 8
    const int r = idx >> 5, c = idx & 31;
    const int hp = r >> 2, vp = r & 3, h = c >> 2, v = c & 3;
    const int m = (hp * h + 2 * vp * v) & 7;
    const float ang = (TWO_PI * 0.125f) * (float)m;
    const float sgn = ((hp + vp) & 1) ? -0.1767766952966369f : 0.1767766952966369f;
    ws[idx]        = sgn * cosf(ang);
    ws[1024 + idx] = sgn * sinf(ang);          // ifft => +i sin
  }
  if (idx < 2304) {  // B[t][k] = (1/sqrt(48)) (-1)^k e^{-2pi i k t/48}, exact phase mod 48
    const int t = idx / 48, k = idx - t * 48;
    const int m = (k * t) % 48;
    const float ang = (TWO_PI / 48.0f) * (float)m;
    const float sgn = (k & 1) ? -0.1443375672974064f : 0.1443375672974064f;
    ws[2048 + idx] =  sgn * cosf(ang);
    ws[4352 + idx] = -sgn * sinf(ang);         // fft => -i sin
  }
}

// One block per (batch, channel). Fused complex transforms:
//   Y(32x96)  = M(32x32) * X(32x96)     (h,v) Kronecker DFT
//   Out(64x48)= Y(64x48) * B(48x48)     time DFT, rows reordered to o = hv*2+p
__global__ __launch_bounds__(256) void beam_fft_wmma_kernel(
    const float* __restrict__ xr_g,
    const float* __restrict__ xi_g,
    const float* __restrict__ tab_g,
    float* __restrict__ out_g) {
  __shared__ __align__(16) float TAB[TAB_FLOATS];   // MR | MI | BR | BI
  __shared__ __align__(16) float XR[32][96];
  __shared__ __align__(16) float XI[32][96];
  __shared__ __align__(16) float YR[64][48];
  __shared__ __align__(16) float YI[64][48];

  float (*MR)[32] = (float(*)[32])&TAB[0];
  float (*MI)[32] = (float(*)[32])&TAB[1024];
  float (*BR)[48] = (float(*)[48])&TAB[2048];
  float (*BI)[48] = (float(*)[48])&TAB[4352];

  const int tid = threadIdx.x;
  const int bc  = blockIdx.x;                       // b*32 + c

  // ---- async-stage X tile (HBM -> LDS, no VGPR round trip; ASYNCcnt tracked) ----
  {
    const unsigned long long srcR =
        (unsigned long long)(uintptr_t)xr_g + (unsigned long long)bc * 3072ull * 4ull;
    const unsigned long long srcI =
        (unsigned long long)(uintptr_t)xi_g + (unsigned long long)bc * 3072ull * 4ull;
    const unsigned ldsR = (unsigned)(uintptr_t)&XR[0][0];
    const unsigned ldsI = (unsigned)(uintptr_t)&XI[0][0];
    for (int i = tid; i < 768; i += 256) {          // 768 x 16B per plane
      const unsigned boff = (unsigned)i * 16u;
      asm volatile("global_load_async_to_lds_b128 %0, %1, %2"
                   :: "v"(ldsR + boff), "v"(boff), "s"(srcR) : "memory");
      asm volatile("global_load_async_to_lds_b128 %0, %1, %2"
                   :: "v"(ldsI + boff), "v"(boff), "s"(srcI) : "memory");
    }
  }

  // ---- pull precomputed DFT tables from L2 into LDS (coalesced, broadcast across blocks) ----
  {
    const v4f* src = (const v4f*)tab_g;
    v4f* dst = (v4f*)&TAB[0];
    for (int i = tid; i < TAB_FLOATS / 4; i += 256) dst[i] = src[i];
  }

  asm volatile("s_wait_asynccnt 0" ::: "memory");
  __syncthreads();

  const int wave = tid >> 5;
  const int lane = tid & 31;
  const int half = lane >> 4;                       // 0: lanes 0-15, 1: lanes 16-31
  const int l    = lane & 15;

  // ---- Stage A: Y = M * X (complex), 2x6 tiles of 16x16, K=32 ----
  for (int tile = wave; tile < 12; tile += 8) {
    const int I = tile / 6, J = tile % 6;
    const int mrow = I * 16 + l;
    const int ncol = J * 16 + l;
    v8f p_rr = {}, p_ii = {}, p_ri = {}, p_ir = {};
    for (int kk = 0; kk < 8; ++kk) {
      const int ka = kk * 4 + half * 2;             // K pair held by this half-wave
      v2f aR = { MR[mrow][ka], MR[mrow][ka + 1] };
      v2f aI = { MI[mrow][ka], MI[mrow][ka + 1] };
      v2f bR = { XR[ka][ncol], XR[ka + 1][ncol] };
      v2f bI = { XI[ka][ncol], XI[ka + 1][ncol] };
      p_rr = WMMA_F32X4(aR, bR, p_rr);
      p_ii = WMMA_F32X4(aI, bI, p_ii);
      p_ri = WMMA_F32X4(aR, bI, p_ri);
      p_ir = WMMA_F32X4(aI, bR, p_ir);
    }
#pragma unroll
    for (int v = 0; v < 8; ++v) {
      const int row = I * 16 + half * 8 + v;        // hv'
      const int col = J * 16 + l;                   // p*48 + t
      const int p = col / 48, t = col - p * 48;
      YR[row * 2 + p][t] = p_rr[v] - p_ii[v];       // stage-B row order o = hv'*2+p
      YI[row * 2 + p][t] = p_ri[v] + p_ir[v];
    }
  }
  __syncthreads();

  // ---- Stage B: Out = Y * B (complex), 4x3 tiles of 16x16, K=48; direct global store ----
  float2* outv = (float2*)out_g;
  const size_t obase = (size_t)bc * 64 * 48;
  for (int tile = wave; tile < 12; tile += 8) {
    const int I = tile / 3, J = tile % 3;
    const int mrow = I * 16 + l;
    const int ncol = J * 16 + l;
    v8f p_rr = {}, p_ii = {}, p_ri = {}, p_ir = {};
    for (int kk = 0; kk < 12; ++kk) {
      const int ka = kk * 4 + half * 2;
      v2f aR = { YR[mrow][ka], YR[mrow][ka + 1] };
      v2f aI = { YI[mrow][ka], YI[mrow][ka + 1] };
      v2f bR = { BR[ka][ncol], BR[ka + 1][ncol] };
      v2f bI = { BI[ka][ncol], BI[ka + 1][ncol] };
      p_rr = WMMA_F32X4(aR, bR, p_rr);
      p_ii = WMMA_F32X4(aI, bI, p_ii);
      p_ri = WMMA_F32X4(aR, bI, p_ri);
      p_ir = WMMA_F32X4(aI, bR, p_ir);
    }
#pragma unroll
    for (int v = 0; v < 8; ++v) {
      const int row = I * 16 + half * 8 + v;        // o = h*8 + v*2 + p
      const int col = J * 16 + l;                   // k (frequency)
      float2 o;
      o.x = p_rr[v] - p_ii[v];
      o.y = p_ri[v] + p_ir[v];
      outv[obase + (size_t)row * 48 + col] = o;
    }
  }
}

extern "C" void kernel_launch(void* const* d_in, const int* in_sizes, int n_in,
                              void* d_out, int out_size, void* d_ws, size_t ws_size,
                              hipStream_t stream) {
  (void)in_sizes; (void)n_in; (void)out_size; (void)ws_size;
  const float* xr = (const float*)d_in[0];
  const float* xi = (const float*)d_in[1];
  float* tab = (float*)d_ws;
  // 1) build DFT tables once per launch (26.6 KB in d_ws, L2-resident)
  build_tables_kernel<<<dim3(9), dim3(256), 0, stream>>>(tab);
  // 2) fused transform: 256 batches * 32 channels = 8192 independent (64x48) tiles
  beam_fft_wmma_kernel<<<dim3(8192), dim3(256), 0, stream>>>(xr, xi, tab, (float*)d_out);
}